// GCN_60086592471617
// MI455X (gfx1250) — compile-verified
//
#include <hip/hip_runtime.h>

#define N_NODES 49984
#define N_EDGES 799744
#define DIM_IN  64
#define WIDTH   256
#define NLAYERS 4
#define M_TILES (N_NODES / 16)   // 3124

typedef __attribute__((ext_vector_type(2))) float v2f;
typedef __attribute__((ext_vector_type(8))) float v8f;
typedef int v4i __attribute__((vector_size(4 * sizeof(int))));

#if defined(__has_builtin)
#if __has_builtin(__builtin_amdgcn_global_load_async_to_lds_b128)
#define HAVE_ASYNC_LDS 1
#endif
#endif

__device__ __forceinline__ float gelu_exact(float x) {
    // exact GELU: 0.5*x*(1+erf(x/sqrt(2)))
    return 0.5f * x * (1.0f + erff(x * 0.70710678118654752440f));
}

// Stage `count4` float4s from global `g` to LDS `l`, cooperatively across 256
// threads. Prefers the CDNA5 async-to-LDS path (ASYNCcnt) when available.
__device__ __forceinline__ void stage_tile(const float4* __restrict__ g,
                                           float4* __restrict__ l,
                                           int count4, int tid)
{
#ifdef HAVE_ASYNC_LDS
    for (int i = tid; i < count4; i += 256) {
        __attribute__((address_space(1))) v4i* gp =
            (__attribute__((address_space(1))) v4i*)(uintptr_t)(g + i);
        __attribute__((address_space(3))) v4i* lp =
            (__attribute__((address_space(3))) v4i*)(uint32_t)(uintptr_t)(l + i);
        __builtin_amdgcn_global_load_async_to_lds_b128(gp, lp, 0, 0);
    }
#if __has_builtin(__builtin_amdgcn_s_wait_asynccnt)
    __builtin_amdgcn_s_wait_asynccnt(0);
#else
    asm volatile("s_wait_asynccnt 0x0" ::: "memory");
#endif
#else
    for (int i = tid; i < count4; i += 256) l[i] = g[i];
#endif
}

// ---------------------------------------------------------------------------
// WMMA GEMM: out[mbase:mbase+16, 0:256] = A[mbase:mbase+16, 0:K] * B (+bias,gelu)
//   B(k,o) = Bm[o*K + k]   (B given transposed: Wf [256,64] or WgT [256,256])
// One block = 256 threads = 8 waves; wave w computes N-tiles {w, w+8}.
// ---------------------------------------------------------------------------
template <int K, bool FUSE_GELU>
__global__ __launch_bounds__(256)
void gemm_wmma_kernel(const float* __restrict__ A,
                      const float* __restrict__ Bm,
                      const float* __restrict__ bias,
                      float* __restrict__ out)
{
    __shared__ float As[16 * K];

    const int mbase = blockIdx.x * 16;
    const int tid   = threadIdx.x;

    // stage the 16 x K fp32 A-tile into LDS (contiguous rows)
    stage_tile(reinterpret_cast<const float4*>(A + (size_t)mbase * K),
               reinterpret_cast<float4*>(As), 16 * K / 4, tid);
    __syncthreads();

    const int wave  = tid >> 5;
    const int lane  = tid & 31;
    const int mrow  = lane & 15;        // A row this lane feeds
    const int khalf = lane >> 4;        // 0: K{0,1}, 1: K{2,3} within a K4 step

    #pragma unroll
    for (int t = 0; t < 2; ++t) {
        const int ntile = wave + 8 * t;
        const int col   = ntile * 16 + (lane & 15);

        v8f acc = {};
        #pragma unroll 4
        for (int kk = 0; kk < K; kk += 4) {
            const int k0 = kk + 2 * khalf;
            v2f a = *reinterpret_cast<const v2f*>(&As[mrow * K + k0]);
            v2f b = *reinterpret_cast<const v2f*>(&Bm[(size_t)col * K + k0]);
            acc = __builtin_amdgcn_wmma_f32_16x16x4_f32(
                      false, a, false, b, (short)0, acc, false, false);
        }

        #pragma unroll
        for (int r = 0; r < 8; ++r) {
            const int m = mbase + r + 8 * khalf;
            float v = acc[r];
            if (bias) v += bias[col];
            if (FUSE_GELU) v = gelu_exact(v);
            out[(size_t)m * WIDTH + col] = v;
        }
    }
}

// ---------------------------------------------------------------------------
// One-time transpose: WgT[l][o][k] = Wg[l][k][o]   (4 x 256 x 256)
// ---------------------------------------------------------------------------
__global__ void transpose_wg_kernel(const float* __restrict__ Wg, float* __restrict__ WgT) {
    int i = blockIdx.x * 256 + threadIdx.x;
    if (i >= NLAYERS * WIDTH * WIDTH) return;
    int l   = i >> 16;
    int rem = i & 0xFFFF;
    int k   = rem >> 8;
    int o   = rem & 255;
    WgT[(size_t)l * WIDTH * WIDTH + (size_t)o * WIDTH + k] = Wg[i];
}

// ---------------------------------------------------------------------------
// gcn_norm helpers (self-loop included: deg starts at 1)
// ---------------------------------------------------------------------------
__global__ void init_deg_kernel(float* __restrict__ deg) {
    int i = blockIdx.x * 256 + threadIdx.x;
    if (i < N_NODES) deg[i] = 1.0f;
}

__global__ void deg_accum_kernel(const int* __restrict__ col, float* __restrict__ deg) {
    int e = blockIdx.x * 256 + threadIdx.x;
    if (e < N_EDGES) unsafeAtomicAdd(&deg[col[e]], 1.0f);
}

__global__ void rsqrt_kernel(float* __restrict__ d) {
    int i = blockIdx.x * 256 + threadIdx.x;
    if (i < N_NODES) d[i] = rsqrtf(d[i]);   // deg >= 1 always
}

__global__ void norm_kernel(const int* __restrict__ row, const int* __restrict__ col,
                            const float* __restrict__ dis, float* __restrict__ nrm) {
    int e = blockIdx.x * 256 + threadIdx.x;
    if (e < N_EDGES) nrm[e] = dis[row[e]] * dis[col[e]];
}

// ---------------------------------------------------------------------------
// agg = h + bg[l] + xl * dis^2  (residual + bias + self-loop message)
// ---------------------------------------------------------------------------
__global__ __launch_bounds__(256)
void agg_init_kernel(const float* __restrict__ h, const float* __restrict__ xl,
                     const float* __restrict__ bg_l, const float* __restrict__ dis,
                     float* __restrict__ agg)
{
    int i = blockIdx.x * 256 + threadIdx.x;
    if (i >= N_NODES * (WIDTH / 4)) return;
    int n  = i >> 6;
    int ch = i & 63;
    float dd = dis[n]; dd *= dd;
    float4 h4 = reinterpret_cast<const float4*>(h)[i];
    float4 x4 = reinterpret_cast<const float4*>(xl)[i];
    float4 b4 = reinterpret_cast<const float4*>(bg_l)[ch];
    float4 o;
    o.x = h4.x + b4.x + x4.x * dd;
    o.y = h4.y + b4.y + x4.y * dd;
    o.z = h4.z + b4.z + x4.z * dd;
    o.w = h4.w + b4.w + x4.w * dd;
    reinterpret_cast<float4*>(agg)[i] = o;
}

// ---------------------------------------------------------------------------
// edge scatter: agg[col[e]] += xl[row[e]] * norm[e]   (per 4-channel chunk)
// ---------------------------------------------------------------------------
__global__ __launch_bounds__(256)
void scatter_kernel(const int* __restrict__ row, const int* __restrict__ col,
                    const float* __restrict__ nrm, const float* __restrict__ xl,
                    float* __restrict__ agg)
{
    long long idx = (long long)blockIdx.x * 256 + threadIdx.x;
    if (idx >= (long long)N_EDGES * (WIDTH / 4)) return;
    int e  = (int)(idx >> 6);
    int ch = (int)(idx & 63);
    int r  = row[e];
    int c  = col[e];
    float nv = nrm[e];
    float4 m = reinterpret_cast<const float4*>(xl)[(size_t)r * 64 + ch];
    float* dst = agg + (size_t)c * WIDTH + ch * 4;
    unsafeAtomicAdd(dst + 0, m.x * nv);
    unsafeAtomicAdd(dst + 1, m.y * nv);
    unsafeAtomicAdd(dst + 2, m.z * nv);
    unsafeAtomicAdd(dst + 3, m.w * nv);
}

// ---------------------------------------------------------------------------
// head: out[g,o] = sum_k gelu(h[node,k]) * Wh[o,k] + bh[o], node = center[g]+ptr[g]
// ---------------------------------------------------------------------------
__global__ void head_kernel(const float* __restrict__ h, const float* __restrict__ Wh,
                            const float* __restrict__ bh, const int* __restrict__ center,
                            const int* __restrict__ ptr, float* __restrict__ out)
{
    int tid = threadIdx.x;
    if (tid >= 32 * 7) return;
    int g = tid / 7;
    int o = tid % 7;
    int node = center[g] + ptr[g];
    const float* hr = h  + (size_t)node * WIDTH;
    const float* wr = Wh + (size_t)o * WIDTH;
    float s = bh[o];
    for (int k = 0; k < WIDTH; ++k) s += gelu_exact(hr[k]) * wr[k];
    out[tid] = s;
}

// ---------------------------------------------------------------------------
extern "C" void kernel_launch(void* const* d_in, const int* in_sizes, int n_in,
                              void* d_out, int out_size, void* d_ws, size_t ws_size,
                              hipStream_t stream)
{
    const float* x      = (const float*)d_in[0];
    const int*   ei     = (const int*)  d_in[1];
    const int*   center = (const int*)  d_in[2];
    const int*   ptr    = (const int*)  d_in[3];
    const float* Wf     = (const float*)d_in[4];
    const float* bf     = (const float*)d_in[5];
    const float* Wg     = (const float*)d_in[6];
    const float* bg     = (const float*)d_in[7];
    const float* Wh     = (const float*)d_in[8];
    const float* bh     = (const float*)d_in[9];

    const int* row  = ei;            // edge_index[0] = source
    const int* colp = ei + N_EDGES;  // edge_index[1] = target

    char* ws = (char*)d_ws;
    const size_t szH = (size_t)N_NODES * WIDTH * sizeof(float);
    float* hA  = (float*)(ws);
    float* hB  = (float*)(ws + szH);
    float* xl  = (float*)(ws + 2 * szH);
    float* dis = (float*)(ws + 3 * szH);
    size_t disBytes = (((size_t)N_NODES * sizeof(float)) + 255) & ~(size_t)255;
    float* nrm = (float*)(ws + 3 * szH + disBytes);
    size_t nrmBytes = (((size_t)N_EDGES * sizeof(float)) + 255) & ~(size_t)255;
    float* WgT = (float*)(ws + 3 * szH + disBytes + nrmBytes);

    // 0) transpose layer weights once: WgT[l][o][k]
    transpose_wg_kernel<<<(NLAYERS * WIDTH * WIDTH + 255) / 256, 256, 0, stream>>>(Wg, WgT);

    // 1) foot: hA = gelu(x @ Wf^T + bf)   [WMMA fp32, fused bias+GELU]
    gemm_wmma_kernel<DIM_IN, true><<<M_TILES, 256, 0, stream>>>(x, Wf, bf, hA);

    // 2) gcn_norm (static across layers)
    init_deg_kernel<<<(N_NODES + 255) / 256, 256, 0, stream>>>(dis);
    deg_accum_kernel<<<(N_EDGES + 255) / 256, 256, 0, stream>>>(colp, dis);
    rsqrt_kernel<<<(N_NODES + 255) / 256, 256, 0, stream>>>(dis);
    norm_kernel<<<(N_EDGES + 255) / 256, 256, 0, stream>>>(row, colp, dis, nrm);

    // 3) GCN blocks with residual
    float* hin  = hA;
    float* hout = hB;
    const int aggBlocks  = (N_NODES * (WIDTH / 4) + 255) / 256;
    const int scatBlocks = (int)(((long long)N_EDGES * (WIDTH / 4) + 255) / 256);
    for (int l = 0; l < NLAYERS; ++l) {
        gemm_wmma_kernel<WIDTH, false><<<M_TILES, 256, 0, stream>>>(
            hin, WgT + (size_t)l * WIDTH * WIDTH, nullptr, xl);
        agg_init_kernel<<<aggBlocks, 256, 0, stream>>>(hin, xl, bg + (size_t)l * WIDTH, dis, hout);
        scatter_kernel<<<scatBlocks, 256, 0, stream>>>(row, colp, nrm, xl, hout);
        float* t = hin; hin = hout; hout = t;
    }

    // 4) head on the 32 center nodes
    head_kernel<<<1, 256, 0, stream>>>(hin, Wh, bh, center, ptr, (float*)d_out);
}